// SinusoidalPositionEmbeddings_86260123173361
// MI455X (gfx1250) — compile-verified
//
#include <hip/hip_runtime.h>
#include <math.h>

// Sinusoidal position embeddings:
//   out[b, c] = sin(t[b] * exp(-ln(1e4) * (c - c%2) / (C-1)))   c even
//             = cos( same arg )                                  c odd
//
// B = 131072, C = 1024 -> 512 MiB fp32 write-once stream: HBM-bound
// (~23 us at 23.3 TB/s on MI455X). Strategy:
//   * per-thread div_term frequencies computed once into registers (v_exp_f32)
//   * native v_sin_f32 / v_cos_f32, arguments pre-scaled to revolutions
//     (|arg| <= 1000/(2pi) ~= 159 < 256 hw valid domain -> no range reduction)
//   * 128-bit non-temporal stores: stream is 2.7x the 192 MB L2, keep it out
//   * compile-time trip count (16 rows/block) -> no remainder loop, minimal salu
// No WMMA/TDM: there is no matrix contraction here and staging stores through
// LDS would only add traffic to a pure store stream.

#define EMBED_C 1024

typedef __attribute__((ext_vector_type(4))) float v4f;

template <int ROWS>
__global__ __launch_bounds__(256) void
sinemb_nt_kernel(const float* __restrict__ t, float* __restrict__ out,
                 int rows_rt, int batch)
{
    const int tid = threadIdx.x;          // 0..255 ; 4 channels per lane
    const int c0  = tid << 2;

    // pair index p = c/2 ; div_term(p) = exp2(-log2(1e4) * 2p / (C-1))
    const float p0   = (float)(c0 >> 1);
    const float kexp = -13.287712379549449f * (2.0f / 1023.0f);
    const float INV_2PI = 0.15915494309189535f;

    const float s0 = __builtin_amdgcn_exp2f(kexp * p0) * INV_2PI;
    const float s1 = __builtin_amdgcn_exp2f(kexp * (p0 + 1.0f)) * INV_2PI;

    // contiguous row range per block: t-values share a cacheline (scalar load
    // path), 64 KB contiguous output stream per block
    const int rows = (ROWS > 0) ? ROWS : rows_rt;
    int r    = blockIdx.x * rows;
    int rend = r + rows;
    if (ROWS <= 0 && rend > batch) rend = batch;   // generic fallback only

    float* p = out + (size_t)r * EMBED_C + (size_t)c0;

    #pragma unroll 4
    for (; r < rend; ++r, p += EMBED_C) {
        const float tv = t[r];            // block-uniform -> s_load
        const float a0 = tv * s0;         // revolutions
        const float a1 = tv * s1;

        v4f v;
        v.x = __builtin_amdgcn_sinf(a0);  // even channel: sin
        v.y = __builtin_amdgcn_cosf(a0);  // odd  channel: cos
        v.z = __builtin_amdgcn_sinf(a1);
        v.w = __builtin_amdgcn_cosf(a1);

        __builtin_nontemporal_store(v, (v4f*)p);   // global_store_b128 ... nt
    }
}

extern "C" void kernel_launch(void* const* d_in, const int* in_sizes, int n_in,
                              void* d_out, int out_size, void* d_ws, size_t ws_size,
                              hipStream_t stream)
{
    (void)n_in; (void)out_size; (void)d_ws; (void)ws_size;
    const float* t   = (const float*)d_in[0];
    float*       out = (float*)d_out;
    const int batch  = in_sizes[0];       // 131072 in the harness

    // 256 threads = 8 wave32 covering one full 1024-channel row per iteration.
    if (batch % 8192 == 0 && batch / 8192 == 16) {
        // harness shape: 8192 blocks x 16 rows, compile-time trip count
        sinemb_nt_kernel<16><<<dim3(8192), dim3(256), 0, stream>>>(t, out, 16, batch);
    } else {
        int blocks = batch < 8192 ? batch : 8192;
        int rpb    = (batch + blocks - 1) / blocks;
        sinemb_nt_kernel<0><<<dim3(blocks), dim3(256), 0, stream>>>(t, out, rpb, batch);
    }
}